// ScaledDotProductAttention_11158325035340
// MI455X (gfx1250) — compile-verified
//
#include <hip/hip_runtime.h>

typedef float v2f __attribute__((ext_vector_type(2)));
typedef float v8f __attribute__((ext_vector_type(8)));

#define B_ 4
#define H_ 16
#define L_ 1024
#define D_ 64
#define QT 16              // query rows per workgroup
#define NWAVE 4            // waves per workgroup (wave32)
#define LDS_STRIDE 1028    // 1024 + 4 floats pad: 1028 % 64 == 4 -> conflict-free column reads

__global__ __launch_bounds__(128) void sdpa_wmma_f32_kernel(
    const float* __restrict__ Q, const float* __restrict__ K,
    const float* __restrict__ V, const int* __restrict__ mask,
    const float* __restrict__ bias,
    float* __restrict__ out, float* __restrict__ score)
{
    __shared__ float sP[QT * LDS_STRIDE];   // score / probability tile 16 x 1024 (padded)
    __shared__ float sM[QT];                // per-row max
    __shared__ float sS[QT];                // per-row 1/sum

    const int tid  = threadIdx.x;
    const int wave = tid >> 5;
    const int lane = tid & 31;
    const int half = lane >> 4;   // 0: K even pair, 1: K odd pair (per 16x4 f32 A/B layout)
    const int l16  = lane & 15;

    const int gid = blockIdx.x;
    const int qt  = gid & 63;          // L_/QT tiles
    const int h   = (gid >> 6) & (H_ - 1);
    const int b   = gid >> 10;

    const int    q0 = qt * QT;
    const size_t bh = (size_t)(b * H_ + h);

    const float* Qb = Q + (bh * L_ + q0) * D_;
    const float* Kb = K + bh * L_ * D_;
    const float* Vb = V + bh * L_ * D_;
    const int*   Mb = mask + ((size_t)b * L_ + q0) * L_;   // mask is [B,1,L,L]
    const float* Bb = bias + ((size_t)h * L_ + q0) * L_;   // bias is [1,H,L,L]

    // ---- preload Q A-fragments: 16 chunks covering K(=D) 0..63 in steps of 4 ----
    // lane layout (A 16x4 f32): row = lane&15; VGPR pair holds K = 4*kc + 2*half + {0,1}
    v2f aq[16];
    {
        const float* qrow = Qb + (size_t)l16 * D_ + 2 * half;
        #pragma unroll
        for (int kc = 0; kc < 16; ++kc)
            aq[kc] = *(const v2f*)(qrow + 4 * kc);
    }

    // ---- S = (Q K^T) * 1/sqrt(64), masked + biased, staged in LDS ----
    for (int nt = wave; nt < L_ / 16; nt += NWAVE) {
        v8f acc = {};
        const float* krow = Kb + (size_t)(nt * 16 + l16) * D_ + 2 * half;
        #pragma unroll
        for (int kc = 0; kc < 16; ++kc) {
            v2f bk = *(const v2f*)(krow + 4 * kc);
            acc = __builtin_amdgcn_wmma_f32_16x16x4_f32(
                false, aq[kc], false, bk, (short)0, acc, false, false);
        }
        // C/D layout: element (row = r + 8*half, col = l16) of the 16x16 tile
        #pragma unroll
        for (int r = 0; r < 8; ++r) {
            const int row = r + 8 * half;
            const int col = nt * 16 + l16;
            float s = acc[r] * 0.125f;                       // 1/sqrt(64)
            const int m = Mb[(size_t)row * L_ + col];
            s = (m == 0) ? -10000.0f : s;
            s += Bb[(size_t)row * L_ + col];
            sP[row * LDS_STRIDE + col] = s;
        }
    }
    __syncthreads();

    // ---- softmax row statistics: 8 lanes cooperate per row ----
    {
        const int row = tid >> 3;       // 0..15
        const int sub = tid & 7;
        const float* prow = sP + row * LDS_STRIDE;
        float m = -__builtin_inff();
        for (int c = sub; c < L_; c += 8) m = fmaxf(m, prow[c]);
        #pragma unroll
        for (int off = 4; off; off >>= 1) m = fmaxf(m, __shfl_xor(m, off, 32));
        float s = 0.0f;
        for (int c = sub; c < L_; c += 8) s += __expf(prow[c] - m);
        #pragma unroll
        for (int off = 4; off; off >>= 1) s += __shfl_xor(s, off, 32);
        if (sub == 0) { sM[row] = m; sS[row] = 1.0f / s; }
    }
    __syncthreads();

    // ---- normalize: write score to HBM (coalesced) and keep P in LDS ----
    {
        float* scoreBase = score + (bh * L_ + q0) * L_;
        for (int idx = tid; idx < QT * L_; idx += 128) {
            const int row = idx >> 10;         // L_ = 1024
            const int col = idx & (L_ - 1);
            const float p = __expf(sP[row * LDS_STRIDE + col] - sM[row]) * sS[row];
            sP[row * LDS_STRIDE + col] = p;
            scoreBase[(size_t)row * L_ + col] = p;
        }
    }
    __syncthreads();

    // ---- out = P @ V : wave `w` owns output D-tile dt = w (16 columns) ----
    {
        const int dt = wave;
        v8f acc = {};
        const float* vcol = Vb + dt * 16 + l16;
        for (int kc = 0; kc < L_ / 4; ++kc) {
            const int kk = 4 * kc + 2 * half;
            // A from LDS (conflict-free: banks = 4*l16 + kk + {0,1}, all distinct mod 64)
            v2f ap = *(const v2f*)(sP + l16 * LDS_STRIDE + kk);
            v2f bv;
            bv.x = vcol[(size_t)kk * D_];
            bv.y = vcol[(size_t)(kk + 1) * D_];
            acc = __builtin_amdgcn_wmma_f32_16x16x4_f32(
                false, ap, false, bv, (short)0, acc, false, false);
        }
        float* ob = out + (bh * L_ + q0) * D_ + dt * 16;
        #pragma unroll
        for (int r = 0; r < 8; ++r) {
            const int row = r + 8 * half;
            ob[(size_t)row * D_ + l16] = acc[r];
        }
    }
}

extern "C" void kernel_launch(void* const* d_in, const int* in_sizes, int n_in,
                              void* d_out, int out_size, void* d_ws, size_t ws_size,
                              hipStream_t stream) {
    (void)in_sizes; (void)n_in; (void)d_ws; (void)ws_size; (void)out_size;
    const float* Q    = (const float*)d_in[0];
    const float* K    = (const float*)d_in[1];
    const float* V    = (const float*)d_in[2];
    const int*   mask = (const int*)d_in[3];
    const float* bias = (const float*)d_in[4];

    float* out   = (float*)d_out;                              // [B,H,L,D]
    float* score = out + (size_t)B_ * H_ * L_ * D_;            // [B,H,L,L]

    const int grid = B_ * H_ * (L_ / QT);                      // 4096 workgroups
    sdpa_wmma_f32_kernel<<<grid, 128, 0, stream>>>(Q, K, V, mask, bias, out, score);
}